// Net_6004364280103
// MI455X (gfx1250) — compile-verified
//
#include <hip/hip_runtime.h>
#include <hip/hip_bf16.h>

typedef __attribute__((ext_vector_type(2))) float v2f;
typedef __attribute__((ext_vector_type(8))) float v8f;

// ---------------- degree / normalization ----------------

__global__ void k_init_deg(float* __restrict__ deg, int n) {
    int i = blockIdx.x * blockDim.x + threadIdx.x;
    if (i < n) deg[i] = 1.0f;           // self-loop contributes 1
}

__global__ void k_deg_accum(const int* __restrict__ dst, float* __restrict__ deg, int E) {
    int i = blockIdx.x * blockDim.x + threadIdx.x;
    if (i < E) atomicAdd(&deg[dst[i]], 1.0f);
}

__global__ void k_rsqrt(const float* __restrict__ deg, float* __restrict__ dis, int n) {
    int i = blockIdx.x * blockDim.x + threadIdx.x;
    if (i < n) dis[i] = rsqrtf(deg[i]); // deg >= 1 always (self-loop)
}

// ---------------- WMMA fp32 GEMM:  H[n,16] = A[n,K] @ W[K,16] ----------------
// One wave32 computes one 16x16 output tile via V_WMMA_F32_16X16X4_F32,
// sweeping K in steps of 4.  A-frag: lane<16 holds K={k,k+1}, lane>=16 holds
// K={k+2,k+3} of row (lane&15).  B-frag mirrors the K split across lane halves.
// C/D: VGPR r -> row (r + 8*(lane>=16)), col (lane&15).

template <int K>
__global__ __launch_bounds__(256) void k_gemm_wmma(const float* __restrict__ A,
                                                   const float* __restrict__ W,
                                                   float* __restrict__ H,
                                                   int nTiles) {
    int wave = (int)((blockIdx.x * blockDim.x + threadIdx.x) >> 5);
    if (wave >= nTiles) return;              // wave-uniform: EXEC all-ones below
    int lane = threadIdx.x & 31;
    int half = lane >> 4;                    // 0: lanes 0-15, 1: lanes 16-31
    int lm   = lane & 15;
    size_t rowBase = (size_t)wave * 16;
    const float* arow = A + (rowBase + (size_t)lm) * (size_t)K;

    v8f c = {};
#pragma unroll 4
    for (int k = 0; k < K; k += 4) {
        int ka = k + 2 * half;
        v2f a, b;
        a.x = arow[ka];
        a.y = arow[ka + 1];
        b.x = W[ka * 16 + lm];
        b.y = W[(ka + 1) * 16 + lm];
        // (neg_a, A, neg_b, B, c_mod, C, reuse_a, reuse_b)
        c = __builtin_amdgcn_wmma_f32_16x16x4_f32(false, a, false, b,
                                                  (short)0, c, false, false);
    }

    float* hrow = H + rowBase * 16;
#pragma unroll
    for (int r = 0; r < 8; ++r) {
        hrow[(size_t)(r + 8 * half) * 16 + lm] = c[r];
    }
}

// ---------------- self-loop seed:  out = h * dis^2 ----------------

__global__ void k_selfloop(const float* __restrict__ h, const float* __restrict__ dis,
                           float* __restrict__ out, int n16) {
    int i = blockIdx.x * blockDim.x + threadIdx.x;
    if (i < n16) {
        float d = dis[i >> 4];
        out[i] = h[i] * d * d;
    }
}

// ---------------- edge scatter:  out[d] += dis[s]*dis[d]*h[s] ----------------
// 4 lanes per edge, each handles a float4 slice of the 16 features.

__global__ void k_edge_scatter(const int* __restrict__ src, const int* __restrict__ dst,
                               const float* __restrict__ dis, const float* __restrict__ h,
                               float* __restrict__ out, int E) {
    int t = blockIdx.x * blockDim.x + threadIdx.x;
    int e = t >> 2;
    if (e >= E) return;
    int q = t & 3;
    int s = src[e];
    int d = dst[e];
    float norm = dis[s] * dis[d];
    const float4* hp = (const float4*)(h + (size_t)s * 16 + (size_t)q * 4);
    float4 v = *hp;
    float* o = out + (size_t)d * 16 + (size_t)q * 4;
    atomicAdd(o + 0, norm * v.x);
    atomicAdd(o + 1, norm * v.y);
    atomicAdd(o + 2, norm * v.z);
    atomicAdd(o + 3, norm * v.w);
}

// ---------------- driver ----------------

extern "C" void kernel_launch(void* const* d_in, const int* in_sizes, int n_in,
                              void* d_out, int out_size, void* d_ws, size_t ws_size,
                              hipStream_t stream) {
    const float* x  = (const float*)d_in[0];   // [N,512]
    const int*   ei = (const int*)d_in[1];     // [2,E]
    const float* w1 = (const float*)d_in[2];   // [512,16]
    const float* w2 = (const float*)d_in[3];   // [16,16]
    float* out = (float*)d_out;                // [N,16]

    const int N = in_sizes[0] / 512;
    const int E = in_sizes[1] / 2;
    const int* src = ei;
    const int* dst = ei + E;

    // workspace layout (floats)
    float* h1   = (float*)d_ws;                       // N*16
    float* agg1 = h1 + (size_t)N * 16;                // N*16
    float* h2   = agg1 + (size_t)N * 16;              // N*16
    float* deg  = h2 + (size_t)N * 16;                // N
    float* dis  = deg + N;                            // N

    const int B = 256;
    int nTiles = (N + 15) / 16;                       // 6250 for N=100000
    int gemmBlocks = (nTiles * 32 + B - 1) / B;

    // normalization
    k_init_deg<<<(N + B - 1) / B, B, 0, stream>>>(deg, N);
    k_deg_accum<<<(E + B - 1) / B, B, 0, stream>>>(dst, deg, E);
    k_rsqrt<<<(N + B - 1) / B, B, 0, stream>>>(deg, dis, N);

    // layer 1
    k_gemm_wmma<512><<<gemmBlocks, B, 0, stream>>>(x, w1, h1, nTiles);
    k_selfloop<<<(N * 16 + B - 1) / B, B, 0, stream>>>(h1, dis, agg1, N * 16);
    k_edge_scatter<<<((size_t)E * 4 + B - 1) / B, B, 0, stream>>>(src, dst, dis, h1, agg1, E);

    // layer 2
    k_gemm_wmma<16><<<gemmBlocks, B, 0, stream>>>(agg1, w2, h2, nTiles);
    k_selfloop<<<(N * 16 + B - 1) / B, B, 0, stream>>>(h2, dis, out, N * 16);
    k_edge_scatter<<<((size_t)E * 4 + B - 1) / B, B, 0, stream>>>(src, dst, dis, h2, out, E);
}